// QueryAndGroup_19550691131908
// MI455X (gfx1250) — compile-verified
//
#include <hip/hip_runtime.h>

typedef __attribute__((ext_vector_type(2))) float v2f;
typedef __attribute__((ext_vector_type(8))) float v8f;

#define BQ_B 8
#define BQ_N 16384
#define BQ_S 2048      // 2^11
#define BQ_C 128       // 2^7
#define NSAMPLE 64     // 2^6
#define R2 0.04f       // radius^2, radius = 0.2
#define INV_RADIUS 5.0f

// ---------------------------------------------------------------------------
// Kernel 1: ball query via V_WMMA_F32_16X16X4_F32.
// One wave (32 lanes) owns 16 centroids. Per 16-point tile, one WMMA produces
// the full 16x16 matrix of squared distances:
//   A[m][:] = [qx, qy, qz, 1]        (16x4)
//   B[:][n] = [-2xx, -2xy, -2xz, |x|^2]  (4x16)
//   C[m][n] = |q_m|^2
//   D = A*B + C = |q|^2 + |x|^2 - 2 q.x
// Then ballots per accumulator VGPR give each row's 16-bit hit mask, and lane
// m (m<16) appends hit indices for row m in ascending order (ctz loop),
// matching the reference's sort-ascending-take-64 + pad-with-first semantics.
// ---------------------------------------------------------------------------
__global__ __launch_bounds__(32) void ball_query_wmma(
    const float* __restrict__ xyz,      // [B,N,3]
    const float* __restrict__ new_xyz,  // [B,S,3]
    int* __restrict__ idx_out)          // [B,S,NSAMPLE]
{
  const int g    = blockIdx.x;        // [0, B*(S/16))
  const int b    = g >> 7;            // S/16 = 128
  const int s0   = (g & 127) << 4;    // first centroid of this group
  const int lane = threadIdx.x & 31;
  const int half = lane >> 4;         // 0: K=0,1 lanes ; 1: K=2,3 lanes
  const int ln   = lane & 15;

  // --- A matrix (constant over the N scan) ---
  const float* q = new_xyz + (size_t)((b * BQ_S + s0 + ln) * 3);
  const float qx = q[0], qy = q[1], qz = q[2];
  v2f A;
  A.x = half ? qz   : qx;
  A.y = half ? 1.0f : qy;

  // --- C accumulator init: C[r][n] = |q_{r + 8*half}|^2 ---
  v8f Cm;
#pragma unroll
  for (int r = 0; r < 8; ++r) {
    const float* qq = new_xyz + (size_t)((b * BQ_S + s0 + r + 8 * half) * 3);
    const float a0 = qq[0], a1 = qq[1], a2 = qq[2];
    Cm[r] = a0 * a0 + a1 * a1 + a2 * a2;
  }

  int cnt      = (lane < 16) ? 0 : NSAMPLE;  // lanes 16-31 never append
  int firstIdx = 0;
  const int rowbase = (b * BQ_S + s0 + ln) << 6;  // *NSAMPLE

  for (int nt = 0; nt < BQ_N; nt += 16) {
    // --- B matrix for this 16-point tile ---
    const float* xp = xyz + (size_t)((b * BQ_N + nt + ln) * 3);
    const float xx = xp[0], xy = xp[1], xz = xp[2];
    v2f Bm;
    if (half == 0) { Bm.x = -2.0f * xx; Bm.y = -2.0f * xy; }
    else           { Bm.x = -2.0f * xz; Bm.y = xx * xx + xy * xy + xz * xz; }

    // D[m][n] = squared distance, EXEC is all-1s here (reconverged).
    v8f D = __builtin_amdgcn_wmma_f32_16x16x4_f32(
        false, A, false, Bm, (short)0, Cm, false, false);

    // Per-VGPR hit ballots (uniform values): low 16 bits = row r, high = r+8.
    const unsigned m0 = __builtin_amdgcn_ballot_w32(D[0] < R2);
    const unsigned m1 = __builtin_amdgcn_ballot_w32(D[1] < R2);
    const unsigned m2 = __builtin_amdgcn_ballot_w32(D[2] < R2);
    const unsigned m3 = __builtin_amdgcn_ballot_w32(D[3] < R2);
    const unsigned m4 = __builtin_amdgcn_ballot_w32(D[4] < R2);
    const unsigned m5 = __builtin_amdgcn_ballot_w32(D[5] < R2);
    const unsigned m6 = __builtin_amdgcn_ballot_w32(D[6] < R2);
    const unsigned m7 = __builtin_amdgcn_ballot_w32(D[7] < R2);

    // Lane m selects its own row's 16-bit mask.
    const int sel = ln & 7;
    unsigned mm = m0;
    mm = (sel == 1) ? m1 : mm;
    mm = (sel == 2) ? m2 : mm;
    mm = (sel == 3) ? m3 : mm;
    mm = (sel == 4) ? m4 : mm;
    mm = (sel == 5) ? m5 : mm;
    mm = (sel == 6) ? m6 : mm;
    mm = (sel == 7) ? m7 : mm;
    unsigned rm = (ln & 8) ? (mm >> 16) : (mm & 0xFFFFu);

    // Ordered append (ascending point id) for this lane's centroid row.
    if (lane < 16 && cnt < NSAMPLE) {
      while (rm && cnt < NSAMPLE) {
        const int bit = __builtin_ctz(rm);
        const int p   = nt + bit;
        if (cnt == 0) firstIdx = p;
        idx_out[rowbase + cnt] = p;
        ++cnt;
        rm &= rm - 1;
      }
    }

    // Uniform early exit once every row has 64 hits.
    if (__builtin_amdgcn_ballot_w32(cnt >= NSAMPLE) == 0xFFFFFFFFu) break;
  }

  // Pad short rows with the first hit (0 if the row had no hits).
  if (lane < 16) {
    for (int k = cnt; k < NSAMPLE; ++k) idx_out[rowbase + k] = firstIdx;
  }
}

// ---------------------------------------------------------------------------
// Kernel 2: gather + concat (pure bandwidth, ~1.1 GB of traffic).
// One thread per gathered feature element (2^27 threads, all power-of-two
// index math). Threads with c<3 additionally emit the normalized xyz triple.
// ---------------------------------------------------------------------------
__global__ __launch_bounds__(256) void group_gather(
    const float* __restrict__ xyz,      // [B,N,3]
    const float* __restrict__ new_xyz,  // [B,S,3]
    const float* __restrict__ feat,     // [B,N,C]
    const int*   __restrict__ idx,      // [B,S,NSAMPLE]
    float* __restrict__ out)            // [B,S,NSAMPLE,3+C]
{
  const unsigned tid = blockIdx.x * 256u + threadIdx.x;  // < 2^27
  const unsigned c   = tid & (BQ_C - 1u);   // 0..127
  const unsigned grp = tid >> 7;            // flat (b,s,j), == idx offset
  const unsigned bs  = grp >> 6;            // flat (b,s)
  const unsigned s   = bs & (BQ_S - 1u);
  const unsigned b   = bs >> 11;

  const unsigned i = (unsigned)idx[grp];

  const float f = feat[((b * BQ_N + i) << 7) + c];
  const unsigned obase = grp * 131u;        // 3 + C
  out[obase + 3u + c] = f;

  if (c < 3u) {
    const float xv = xyz[(b * BQ_N + i) * 3u + c];
    const float qv = new_xyz[(b * BQ_S + s) * 3u + c];
    out[obase + c] = (xv - qv) * INV_RADIUS;
  }
}

extern "C" void kernel_launch(void* const* d_in, const int* in_sizes, int n_in,
                              void* d_out, int out_size, void* d_ws, size_t ws_size,
                              hipStream_t stream) {
  (void)in_sizes; (void)n_in; (void)out_size; (void)ws_size;
  const float* xyz     = (const float*)d_in[0];
  const float* new_xyz = (const float*)d_in[1];
  const float* feat    = (const float*)d_in[2];
  float* out = (float*)d_out;
  int* idx_ws = (int*)d_ws;   // 8*2048*64*4 = 4 MB

  // Phase 1: ball query. One wave per 16 centroids.
  ball_query_wmma<<<BQ_B * (BQ_S / 16), 32, 0, stream>>>(xyz, new_xyz, idx_ws);

  // Phase 2: gather. 2^27 threads.
  const unsigned total = (unsigned)BQ_B * BQ_S * NSAMPLE * BQ_C;  // 134,217,728
  group_gather<<<total / 256u, 256, 0, stream>>>(xyz, new_xyz, feat, idx_ws, out);
}